// myConv_30167850287318
// MI455X (gfx1250) — compile-verified
//
#include <hip/hip_runtime.h>
#include <stdint.h>

// CDNA5 / gfx1250: depthwise 3x3 conv (single shared kernel, zero pad 1)
// Banded-matrix WMMA:  Out(16x16) = sum_dy A_dy(16x18) x B_dy(18x16)
// via V_WMMA_F32_16X16X4_F32 (K padded 18 -> 20, 5 chunks of 4).
// Interior tiles stage their 18x18 halo with GLOBAL_LOAD_ASYNC_TO_LDS_B64
// (ASYNCcnt-tracked, no VGPR round trip); border tiles use a checked path.

typedef __attribute__((ext_vector_type(2))) float v2f;
typedef __attribute__((ext_vector_type(8))) float v8f;

#define IMG_H 224
#define IMG_W 224
#define TILES_X 14            // 224 / 16
#define TILES_PER_PLANE 196   // 14 * 14
#define PLANES 1024           // 16 batch * 64 channels
#define LDS_ROWS 18           // 16 + 2 halo rows
#define LDS_COLS 20           // 16 + 2 halo cols, padded to 20 for K-chunks of 4
#define LDS_TILE (LDS_ROWS * LDS_COLS)  // 360 floats (1440 B, 16B-multiple)
#define WAVES_PER_WG 8

// banded-matrix entry: w[d] for d in {0,1,2}, else 0 (select chain, no spills)
__device__ __forceinline__ float band(int d, float w0, float w1, float w2) {
    float r = 0.0f;
    r = (d == 0) ? w0 : r;
    r = (d == 1) ? w1 : r;
    r = (d == 2) ? w2 : r;
    return r;
}

__global__ __launch_bounds__(256) void conv3x3_wmma_kernel(
        const float* __restrict__ X,
        const float* __restrict__ Wk,
        float* __restrict__ Out)
{
    __shared__ __align__(16) float lds[WAVES_PER_WG][LDS_TILE];

    const int tid  = threadIdx.x;
    const int wave = tid >> 5;
    const int lane = tid & 31;

    // one 16x16 output tile per wave; grid sized exactly -> EXEC all-1s at WMMA
    const int tile  = blockIdx.x * WAVES_PER_WG + wave;
    const int plane = tile / TILES_PER_PLANE;
    const int trem  = tile - plane * TILES_PER_PLANE;
    const int ty    = trem / TILES_X;
    const int tx    = trem - ty * TILES_X;
    const int y0    = ty * 16;
    const int x0    = tx * 16;

    const float* __restrict__ Xp = X   + (size_t)plane * (IMG_H * IMG_W);
    float* __restrict__       Op = Out + (size_t)plane * (IMG_H * IMG_W);

    float* ltile = lds[wave];

    const bool interior = (ty > 0) && (ty < TILES_X - 1) && (tx > 0) && (tx < TILES_X - 1);

    if (interior) {
        // K-pad columns 18,19 must be zero (stale LDS could be NaN; NaN*0 = NaN)
        if (lane < LDS_ROWS) {
            ltile[lane * LDS_COLS + 18] = 0.0f;
            ltile[lane * LDS_COLS + 19] = 0.0f;
        }
        // async-stage 18 rows x 9 8-byte chunks = 162 chunks, global -> LDS
        const float* gbase = Xp + (y0 - 1) * IMG_W + (x0 - 1);
        for (int i = lane; i < LDS_ROWS * 9; i += 32) {
            const int r  = i / 9;
            const int c2 = i - r * 9;                      // 2-float chunk in row
            const unsigned lds_addr =
                (unsigned)(uintptr_t)(ltile + r * LDS_COLS + c2 * 2);
            const unsigned long long gaddr =
                (unsigned long long)(uintptr_t)(gbase + r * IMG_W + c2 * 2);
            asm volatile("global_load_async_to_lds_b64 %0, %1, off"
                         :: "v"(lds_addr), "v"(gaddr) : "memory");
        }
    } else {
        // border: checked staging with zero padding
        for (int i = lane; i < LDS_TILE; i += 32) {
            const int r  = i / LDS_COLS;
            const int c  = i - r * LDS_COLS;
            const int gy = y0 - 1 + r;
            const int gx = x0 - 1 + c;
            float v = 0.0f;
            if (c < 18 && (unsigned)gy < (unsigned)IMG_H && (unsigned)gx < (unsigned)IMG_W)
                v = Xp[gy * IMG_W + gx];
            ltile[i] = v;
        }
    }

    // drain async global->LDS transfers (no-op for border waves: counter == 0)
    asm volatile("s_wait_asynccnt 0x0" ::: "memory");

    // ---- 3x3 weights (wave-uniform) ----
    const float w00 = Wk[0], w01 = Wk[1], w02 = Wk[2];
    const float w10 = Wk[3], w11 = Wk[4], w12 = Wk[5];
    const float w20 = Wk[6], w21 = Wk[7], w22 = Wk[8];

    const int half = lane >> 4;   // 0: K rows {0,1}, D rows 0..7 ; 1: {2,3}, 8..15
    const int l16  = lane & 15;   // A: M index, B/D: N index
    const int j0   = half * 2;    // K sub-index within a chunk of 4

    v8f accA = {};                // dy = 0, 2
    v8f accB = {};                // dy = 1

    #pragma unroll
    for (int dy = 0; dy < 3; ++dy) {
        const float wd0 = (dy == 0) ? w00 : (dy == 1) ? w10 : w20;
        const float wd1 = (dy == 0) ? w01 : (dy == 1) ? w11 : w21;
        const float wd2 = (dy == 0) ? w02 : (dy == 1) ? w12 : w22;
        const float* arow = ltile + (l16 + dy) * LDS_COLS;
        #pragma unroll
        for (int g = 0; g < 5; ++g) {
            const int k0 = 4 * g + j0;
            // A fragment: A[M = l16][K = k0, k0+1]  (16x4 f32 layout)
            v2f a;
            a.x = arow[k0];
            a.y = arow[k0 + 1];
            // B fragment: B[k][n] = w[dy][k - n] on the 3-wide band
            const int d0 = k0 - l16;
            v2f b;
            b.x = band(d0,     wd0, wd1, wd2);
            b.y = band(d0 + 1, wd0, wd1, wd2);
            if (dy == 1)
                accB = __builtin_amdgcn_wmma_f32_16x16x4_f32(
                    false, a, false, b, (short)0, accB, false, false);
            else
                accA = __builtin_amdgcn_wmma_f32_16x16x4_f32(
                    false, a, false, b, (short)0, accA, false, false);
        }
    }

    // ---- store D: VGPR r -> row M = r + 8*half, col N = l16 (coalesced) ----
    float* orow = Op + (y0 + 8 * half) * IMG_W + x0 + l16;
    #pragma unroll
    for (int r = 0; r < 8; ++r)
        orow[r * IMG_W] = accA[r] + accB[r];
}

extern "C" void kernel_launch(void* const* d_in, const int* in_sizes, int n_in,
                              void* d_out, int out_size, void* d_ws, size_t ws_size,
                              hipStream_t stream) {
    (void)in_sizes; (void)n_in; (void)out_size; (void)d_ws; (void)ws_size;
    const float* X  = (const float*)d_in[0];
    const float* Wk = (const float*)d_in[1];
    float*       O  = (float*)d_out;

    const int total_tiles = PLANES * TILES_PER_PLANE;          // 200704
    const int blocks      = total_tiles / WAVES_PER_WG;        // 25088
    conv3x3_wmma_kernel<<<blocks, 256, 0, stream>>>(X, Wk, O);
}